// SSIMModule_15994458211273
// MI455X (gfx1250) — compile-verified
//
#include <hip/hip_runtime.h>
#include <cstdint>

// SSIM (3x3 box window, zero-pad, /9) over 32x3x512x512 fp32, mean clipped to [0,1].
// Memory-bound: ~201 MB @ 23.3 TB/s ~= 8.6us floor. Strategy:
//   - 64x64 output tiles + 1 halo staged into LDS via global_load_async_to_lds_b128
//   - fused 9-tap sums of (x, y, x^2, y^2, xy) from LDS, SSIM per pixel
//   - deterministic 2-stage reduction; final stage uses V_WMMA_F32_16X16X4_F32

#define TILE    64
#define LROWS   66
#define LSTRIDE 72           // [3 pad][halo @3][64 interior @4..67, 16B aligned][halo @68][3 pad]
#define NBLOCKS (96 * 8 * 8) // 6144 partials -> 96 chunks of 64 for the WMMA reduce

typedef __attribute__((ext_vector_type(2))) float v2f;
typedef __attribute__((ext_vector_type(8))) float v8f;

__global__ __launch_bounds__(256)
void ssim_map_kernel(const float* __restrict__ x, const float* __restrict__ y,
                     float* __restrict__ part) {
  __shared__ float lx[LROWS * LSTRIDE];
  __shared__ float ly[LROWS * LSTRIDE];
  __shared__ float red[256];

  const int tid   = threadIdx.x;
  const int plane = blockIdx.y;                 // 0..95  (n*3 + c)
  const int tr    = (int)(blockIdx.x >> 3) * TILE;
  const int tc    = (int)(blockIdx.x & 7) * TILE;

  const float* xp = x + (size_t)plane * 512 * 512;
  const float* yp = y + (size_t)plane * 512 * 512;

  const unsigned lxbase = (unsigned)(uintptr_t)&lx[0];
  const unsigned lybase = (unsigned)(uintptr_t)&ly[0];

  // ---- Stage interior columns (gc = tc..tc+63, always in-bounds) as 16B quads ----
  for (int i = tid; i < LROWS * 16; i += 256) {
    int r  = i >> 4;                    // 0..65 -> global row tr + r - 1
    int q  = i & 15;                    // quad within the 64-wide interior
    int gr = tr + r - 1;
    int li = r * LSTRIDE + 4 + q * 4;   // 16B-aligned LDS float index
    if ((unsigned)gr < 512u) {
      unsigned long long gx = (unsigned long long)(uintptr_t)(xp + (size_t)gr * 512 + tc + q * 4);
      unsigned long long gy = (unsigned long long)(uintptr_t)(yp + (size_t)gr * 512 + tc + q * 4);
      unsigned lax = lxbase + (unsigned)(li * 4);
      unsigned lay = lybase + (unsigned)(li * 4);
      asm volatile("global_load_async_to_lds_b128 %0, %1, off" :: "v"(lax), "v"(gx) : "memory");
      asm volatile("global_load_async_to_lds_b128 %0, %1, off" :: "v"(lay), "v"(gy) : "memory");
    } else {
      float4 z = {0.f, 0.f, 0.f, 0.f};  // zero-padding rows above/below the image
      *(float4*)&lx[li] = z;
      *(float4*)&ly[li] = z;
    }
  }
  // ---- Stage halo columns (gc = tc-1 and tc+64) ----
  for (int i = tid; i < LROWS * 2; i += 256) {
    int r    = i >> 1;
    int side = i & 1;
    int gr   = tr + r - 1;
    int gc   = side ? (tc + 64) : (tc - 1);
    int li   = r * LSTRIDE + (side ? 68 : 3);
    if ((unsigned)gr < 512u && (unsigned)gc < 512u) {
      unsigned long long gx = (unsigned long long)(uintptr_t)(xp + (size_t)gr * 512 + gc);
      unsigned long long gy = (unsigned long long)(uintptr_t)(yp + (size_t)gr * 512 + gc);
      unsigned lax = lxbase + (unsigned)(li * 4);
      unsigned lay = lybase + (unsigned)(li * 4);
      asm volatile("global_load_async_to_lds_b32 %0, %1, off" :: "v"(lax), "v"(gx) : "memory");
      asm volatile("global_load_async_to_lds_b32 %0, %1, off" :: "v"(lay), "v"(gy) : "memory");
    } else {
      lx[li] = 0.f;
      ly[li] = 0.f;
    }
  }
  asm volatile("s_wait_asynccnt 0x0" ::: "memory");  // drain this wave's async-LDS writes
  __syncthreads();

  // ---- Fused 3x3 sums + SSIM; 16 pixels per thread ----
  const float inv9 = 1.0f / 9.0f;
  const float C1 = 1e-4f, C2 = 9e-4f;
  float acc = 0.0f;
  #pragma unroll
  for (int it = 0; it < 16; ++it) {
    int p = it * 256 + tid;
    int r = p >> 6;             // 0..63
    int c = p & 63;             // 0..63 ; center at storage (r+1, c+4)
    int base = r * LSTRIDE + (c + 3);
    float s1 = 0.f, s2 = 0.f, s11 = 0.f, s22 = 0.f, s12 = 0.f;
    #pragma unroll
    for (int dr = 0; dr < 3; ++dr) {
      int o = base + dr * LSTRIDE;
      #pragma unroll
      for (int dc = 0; dc < 3; ++dc) {
        float a = lx[o + dc];
        float b = ly[o + dc];
        s1 += a; s2 += b;
        s11 = fmaf(a, a, s11);
        s22 = fmaf(b, b, s22);
        s12 = fmaf(a, b, s12);
      }
    }
    float mux = s1 * inv9, muy = s2 * inv9;
    float sx  = s11 * inv9 - mux * mux;
    float sy  = s22 * inv9 - muy * muy;
    float sxy = s12 * inv9 - mux * muy;
    float n = (2.0f * mux * muy + C1) * (2.0f * sxy + C2);
    float d = (mux * mux + muy * muy + C1) * (sx + sy + C2);
    acc += n / (d + 1e-12f);
  }

  // ---- Deterministic in-block tree reduction ----
  red[tid] = acc;
  __syncthreads();
  for (int s = 128; s > 0; s >>= 1) {
    if (tid < s) red[tid] += red[tid + s];
    __syncthreads();
  }
  if (tid == 0) part[blockIdx.y * 64 + blockIdx.x] = red[0];
}

// Single-wave final reduction: chain V_WMMA_F32_16X16X4_F32 with B = ones so each
// issue row-sums a 64-partial chunk into the f32 accumulator (fixed order -> deterministic).
__global__ __launch_bounds__(32)
void ssim_final_kernel(const float* __restrict__ part, float* __restrict__ out,
                       int nchunk, float invN) {
  const int lane = threadIdx.x;
  const int m  = lane & 15;
  const int kb = (lane < 16) ? 0 : 2;   // A 16x4 f32 layout: lanes 0-15 K=0,1 ; 16-31 K=2,3
  v2f b; b.x = 1.0f; b.y = 1.0f;        // B = all ones
  v8f c = {};
  for (int ch = 0; ch < nchunk; ++ch) {
    const float* p = part + ch * 64 + m * 4 + kb;
    v2f a; a.x = p[0]; a.y = p[1];
    c = __builtin_amdgcn_wmma_f32_16x16x4_f32(
        /*neg_a=*/false, a, /*neg_b=*/false, b,
        /*c_mod=*/(short)0, c, /*reuse_a=*/false, /*reuse_b=*/false);
  }
  // D col 0: lane 0 holds rows 0-7 in c[0..7], lane 16 holds rows 8-15.
  float rs = c[0] + c[1] + c[2] + c[3] + c[4] + c[5] + c[6] + c[7];
  float hi = __shfl(rs, 16, 32);
  if (lane == 0) {
    float mean = (rs + hi) * invN;
    out[0] = fminf(fmaxf(mean, 0.0f), 1.0f);
  }
}

extern "C" void kernel_launch(void* const* d_in, const int* in_sizes, int n_in,
                              void* d_out, int out_size, void* d_ws, size_t ws_size,
                              hipStream_t stream) {
  const float* x = (const float*)d_in[0];
  const float* y = (const float*)d_in[1];
  float* part = (float*)d_ws;           // 6144 floats, fully overwritten every call

  dim3 grid(64, 96, 1);                 // 8x8 tiles per plane, 96 planes
  ssim_map_kernel<<<grid, 256, 0, stream>>>(x, y, part);

  const float invN = 1.0f / (32.0f * 3.0f * 512.0f * 512.0f);
  ssim_final_kernel<<<1, 32, 0, stream>>>(part, (float*)d_out, NBLOCKS / 64, invN);
}